// Maj3_6949257085013
// MI455X (gfx1250) — compile-verified
//
#include <hip/hip_runtime.h>

typedef __attribute__((ext_vector_type(2))) float v2f;
typedef __attribute__((ext_vector_type(8))) float v8f;

#define C_IN 64
#define C_OUT 64
#define W_IN 32
#define H_IN 32
#define JLEN 576          // c_in * 9
#define GCNT 192          // JLEN / 3 (independent clamp groups)
#define ROWS 16           // output pixels per workgroup
#define NTHREADS 128      // 4 waves; one 16-channel tile per wave
#define PROW 772          // LDS row stride in floats (mod 64 == 4, 16B aligned)
#define WQ_BYTES (C_OUT * GCNT * 4 * sizeof(float))   // 196,608

// Rewrites weights into K-pair layout: wq[ch][g] = {w[j=g], w[j=192+g], w[j=384+g], 0}
__global__ __launch_bounds__(256)
void Maj3_prep_w(const float* __restrict__ w, float* __restrict__ wq) {
    int idx = blockIdx.x * 256 + threadIdx.x;       // over 64*192
    if (idx < C_OUT * GCNT) {
        int ch = idx / GCNT;
        int g  = idx - ch * GCNT;
        const float* src = w + ch * JLEN;
        float4 v = make_float4(src[g], src[GCNT + g], src[2 * GCNT + g], 0.0f);
        *(float4*)(wq + (size_t)idx * 4) = v;
    }
}

template <bool USEWQ>
__global__ __launch_bounds__(NTHREADS)
void Maj3_kernel(const float* __restrict__ x,
                 const float* __restrict__ w,
                 const float* __restrict__ wq,
                 float* __restrict__ out) {
    // Pair layout: P[r][g] = {k0, k1, k2, 0}; lo lanes read bytes 0-7, hi 8-15
    __shared__ __align__(16) float P[ROWS * PROW];   // ~49.4 KB

    const int tid  = threadIdx.x;
    const int wave = tid >> 5;
    const int lane = tid & 31;
    const int n0   = blockIdx.x * ROWS;

    // ---- Stage patch tile as K-quads (materialize -1 spatial padding) ----
    for (int idx = tid; idx < ROWS * GCNT; idx += NTHREADS) {
        int r = idx / GCNT;
        int g = idx - r * GCNT;
        int n = n0 + r;
        int h  = n & (H_IN - 1);
        int wqd = (n >> 5) & (W_IN - 1);
        int b  = n >> 10;
        float vals[3];
        #pragma unroll
        for (int u = 0; u < 3; ++u) {
            int j  = u * GCNT + g;       // j = c*9 + kw*3 + kh
            int c  = j / 9;
            int rr = j - c * 9;
            int kw = rr / 3;
            int kh = rr - kw * 3;
            int wi = wqd + kw - 1;
            int hi = h + kh - 1;
            float v = -1.0f;
            if (wi >= 0 && wi < W_IN && hi >= 0 && hi < H_IN)
                v = x[((b * W_IN + wi) * H_IN + hi) * C_IN + c];
            vals[u] = v;
        }
        *(float4*)(P + r * PROW + g * 4) = make_float4(vals[0], vals[1], vals[2], 0.0f);
    }
    __syncthreads();

    // ---- Branchless per-lane pair pointers ----
    // A 16x4 f32: lanes 0-15 supply K=0/1, lanes 16-31 supply K=2/3(=0)
    const bool lo   = lane < 16;
    const int  rowm = lane & 15;
    const int  chm  = wave * 16 + rowm;
    const int  half = lo ? 0 : 2;
    const float* pA = P + rowm * PROW + half;                        // LDS pairs
    const float* pW = wq + ((size_t)chm * GCNT) * 4 + half;          // global pairs (USEWQ)
    const float* wbase = w + chm * JLEN;                             // fallback streams
    const float* wp0 = wbase + (lo ? 0 : 384);
    const float* wp1 = wbase + (lo ? 192 : 384);  // dummy for hi lanes; A.y==0 kills k=3

    v8f acc = {};
    if (USEWQ) {
        #pragma unroll 4
        for (int g = 0; g < GCNT; ++g) {
            v2f a  = *(const v2f*)(pA + g * 4);      // ds_load_b64 -> even pair
            v2f bm = *(const v2f*)(pW + g * 4);      // global_load_b64 -> even pair
            v8f zero = {};
            v8f d = __builtin_amdgcn_wmma_f32_16x16x4_f32(
                        false, a, false, bm, (short)0, zero, false, false);
            #pragma unroll
            for (int i = 0; i < 8; ++i)
                acc[i] += __builtin_amdgcn_fmed3f(d[i], -1.0f, 1.0f);
        }
    } else {
        for (int g = 0; g < GCNT; g += 4) {
            float4 bx = *(const float4*)(wp0 + g);
            float4 by = *(const float4*)(wp1 + g);
            #pragma unroll
            for (int q = 0; q < 4; ++q) {
                v2f a = *(const v2f*)(pA + (g + q) * 4);
                v2f bm;
                bm.x = (q == 0) ? bx.x : (q == 1) ? bx.y : (q == 2) ? bx.z : bx.w;
                bm.y = (q == 0) ? by.x : (q == 1) ? by.y : (q == 2) ? by.z : by.w;
                v8f zero = {};
                v8f d = __builtin_amdgcn_wmma_f32_16x16x4_f32(
                            false, a, false, bm, (short)0, zero, false, false);
                #pragma unroll
                for (int i = 0; i < 8; ++i)
                    acc[i] += __builtin_amdgcn_fmed3f(d[i], -1.0f, 1.0f);
            }
        }
    }

    // ---- Writeback: C/D 16x16 layout: row = i + (lane<16?0:8), col = lane&15
    #pragma unroll
    for (int i = 0; i < 8; ++i) {
        int m = i + (lo ? 0 : 8);
        out[(n0 + m) * C_OUT + chm] = acc[i] * 3.0f;
    }
}

extern "C" void kernel_launch(void* const* d_in, const int* in_sizes, int n_in,
                              void* d_out, int out_size, void* d_ws, size_t ws_size,
                              hipStream_t stream) {
    const float* x = (const float*)d_in[0];      // (8,32,32,64) f32
    const float* w = (const float*)d_in[1];      // (64,576) f32 (+/-1)
    float* out = (float*)d_out;                  // (8,32,32,64) f32
    (void)in_sizes; (void)n_in; (void)out_size;

    const int nPix = 8 * W_IN * H_IN;            // 8192
    dim3 grid(nPix / ROWS), block(NTHREADS);

    if (ws_size >= WQ_BYTES) {
        float* wqb = (float*)d_ws;
        Maj3_prep_w<<<dim3((C_OUT * GCNT + 255) / 256), dim3(256), 0, stream>>>(w, wqb);
        Maj3_kernel<true><<<grid, block, 0, stream>>>(x, w, wqb, out);
    } else {
        Maj3_kernel<false><<<grid, block, 0, stream>>>(x, w, nullptr, out);
    }
}